// SCAESuite_9311489097976
// MI455X (gfx1250) — compile-verified
//
#include <hip/hip_runtime.h>

// Problem constants (match reference)
#define B_  4
#define S_  512
#define D_  768
#define H_  12
#define DH_ 64
#define F_  2048
#define BS_ (B_*S_)

typedef float v2f __attribute__((ext_vector_type(2)));
typedef float v8f __attribute__((ext_vector_type(8)));

// ---------------- WMMA f32 tiled GEMM ----------------
// C[M,N] (+)= A[M,K] * B[K,N]      (transB=0, Bmem row-major, ldb = N-stride)
// C[M,N] (+)= A[M,K] * Bmem^T      (transB=1, Bmem is [N,K] row-major, ldb = K-stride)
// Requires M%128==0, N%128==0, K%32==0. 256 threads = 8 waves (wave32).
// LDS is stored K-pair interleaved (float2) so each WMMA fragment is one
// aligned ds_load_b64 directly into an even VGPR pair (no repack moves).
#define BM 128
#define BN 128
#define BK 32
#define KP 16          // BK/2 k-pairs
#define LSTR 144       // padded row stride in float2 units: 2*144 mod 64 = 32
                       // -> half-wave (lane>>4) k-groups hit disjoint bank halves

__global__ __launch_bounds__(256) void gemm_f32_wmma(
    const float* __restrict__ A, const float* __restrict__ Bm,
    float* __restrict__ C, int M, int N, int K,
    int lda, int ldb, int ldc, int transB, int accum)
{
  __shared__ float2 As2[KP][LSTR];
  __shared__ float2 Bs2[KP][LSTR];

  const int tid  = threadIdx.x;
  const int lane = tid & 31;
  const int hl   = lane >> 4;     // half-wave selector (K split)
  const int l15  = lane & 15;
  const int wv   = tid >> 5;      // 0..7
  const int wm   = wv >> 1;       // 0..3 : 32-row slab (2 x 16)
  const int wn   = wv & 1;        // 0..1 : 64-col slab (4 x 16)
  const int M0 = blockIdx.y * BM;
  const int N0 = blockIdx.x * BN;

  v8f acc00 = {}; v8f acc01 = {}; v8f acc02 = {}; v8f acc03 = {};
  v8f acc10 = {}; v8f acc11 = {}; v8f acc12 = {}; v8f acc13 = {};

  // staging thread maps
  const int am  = tid >> 1;           // 0..127 : A row (or B^T n-row)
  const int ak  = (tid & 1) * 16;     // k segment base
  const int bk2 = tid >> 4;           // 0..15  : plain-B k-pair
  const int bnq = (tid & 15) * 4;     // n quad

  float4 ra[4], rb[4];

  auto loadA = [&](int k0) {
    const float* pA = A + (size_t)(M0 + am) * lda + k0 + ak;
    ra[0] = *(const float4*)(pA + 0);
    ra[1] = *(const float4*)(pA + 4);
    ra[2] = *(const float4*)(pA + 8);
    ra[3] = *(const float4*)(pA + 12);
  };
  auto loadB = [&](int k0) {
    if (!transB) {
      const float* pB = Bm + (size_t)(k0 + 2 * bk2) * ldb + N0 + bnq;
      rb[0] = *(const float4*)(pB);
      rb[1] = *(const float4*)(pB + ldb);
      rb[2] = *(const float4*)(pB + 64);
      rb[3] = *(const float4*)(pB + ldb + 64);
    } else {
      const float* pB = Bm + (size_t)(N0 + am) * ldb + k0 + ak;
      rb[0] = *(const float4*)(pB + 0);
      rb[1] = *(const float4*)(pB + 4);
      rb[2] = *(const float4*)(pB + 8);
      rb[3] = *(const float4*)(pB + 12);
    }
  };
  auto storeA = [&]() {
    #pragma unroll
    for (int i = 0; i < 4; ++i) {
      const int kp = (ak + 4 * i) >> 1;
      As2[kp + 0][am] = make_float2(ra[i].x, ra[i].y);
      As2[kp + 1][am] = make_float2(ra[i].z, ra[i].w);
    }
  };
  auto storeB = [&]() {
    if (!transB) {
      Bs2[bk2][bnq + 0]  = make_float2(rb[0].x, rb[1].x);
      Bs2[bk2][bnq + 1]  = make_float2(rb[0].y, rb[1].y);
      Bs2[bk2][bnq + 2]  = make_float2(rb[0].z, rb[1].z);
      Bs2[bk2][bnq + 3]  = make_float2(rb[0].w, rb[1].w);
      Bs2[bk2][bnq + 64] = make_float2(rb[2].x, rb[3].x);
      Bs2[bk2][bnq + 65] = make_float2(rb[2].y, rb[3].y);
      Bs2[bk2][bnq + 66] = make_float2(rb[2].z, rb[3].z);
      Bs2[bk2][bnq + 67] = make_float2(rb[2].w, rb[3].w);
    } else {
      #pragma unroll
      for (int i = 0; i < 4; ++i) {
        const int kp = (ak + 4 * i) >> 1;
        Bs2[kp + 0][am] = make_float2(rb[i].x, rb[i].y);
        Bs2[kp + 1][am] = make_float2(rb[i].z, rb[i].w);
      }
    }
  };

  // prologue: stage 0 into LDS
  loadA(0); loadB(0);
  storeA(); storeB();
  __syncthreads();

  const int nstage = K / BK;
  const int mrow = wm * 32 + l15;
  const int ncol = wn * 64 + l15;

  for (int s = 0; s < nstage; ++s) {
    // pipeline: pull stage s+1 into registers while computing stage s
    if (s + 1 < nstage) { loadA((s + 1) * BK); loadB((s + 1) * BK); }
    // and prefetch stage s+2 into cache (global_prefetch_b8)
    if (s + 2 < nstage) {
      const int k2 = (s + 2) * BK;
      __builtin_prefetch(A + (size_t)(M0 + am) * lda + k2 + ak, 0, 1);
      if (!transB) __builtin_prefetch(Bm + (size_t)(k2 + 2 * bk2) * ldb + N0 + bnq, 0, 1);
      else         __builtin_prefetch(Bm + (size_t)(N0 + am) * ldb + k2 + ak, 0, 1);
    }

    // ISA 16x16x4 f32 frags: lane holds (m=lane%16, k=2*(lane/16)+j) for A,
    // (k=2*(lane/16)+j, n=lane%16) for B -> k-pair index kp = 2*kk + (lane>>4)
    #pragma unroll
    for (int kk = 0; kk < 8; ++kk) {
      const int kp = kk * 2 + hl;
      v2f a0 = *(const v2f*)&As2[kp][mrow];
      v2f a1 = *(const v2f*)&As2[kp][mrow + 16];
      v2f b0 = *(const v2f*)&Bs2[kp][ncol];
      v2f b1 = *(const v2f*)&Bs2[kp][ncol + 16];
      v2f b2 = *(const v2f*)&Bs2[kp][ncol + 32];
      v2f b3 = *(const v2f*)&Bs2[kp][ncol + 48];
      acc00 = __builtin_amdgcn_wmma_f32_16x16x4_f32(false, a0, false, b0, (short)0, acc00, false, false);
      acc01 = __builtin_amdgcn_wmma_f32_16x16x4_f32(false, a0, false, b1, (short)0, acc01, false, false);
      acc02 = __builtin_amdgcn_wmma_f32_16x16x4_f32(false, a0, false, b2, (short)0, acc02, false, false);
      acc03 = __builtin_amdgcn_wmma_f32_16x16x4_f32(false, a0, false, b3, (short)0, acc03, false, false);
      acc10 = __builtin_amdgcn_wmma_f32_16x16x4_f32(false, a1, false, b0, (short)0, acc10, false, false);
      acc11 = __builtin_amdgcn_wmma_f32_16x16x4_f32(false, a1, false, b1, (short)0, acc11, false, false);
      acc12 = __builtin_amdgcn_wmma_f32_16x16x4_f32(false, a1, false, b2, (short)0, acc12, false, false);
      acc13 = __builtin_amdgcn_wmma_f32_16x16x4_f32(false, a1, false, b3, (short)0, acc13, false, false);
    }
    __syncthreads();
    if (s + 1 < nstage) { storeA(); storeB(); __syncthreads(); }
  }

  // C layout: VGPR r -> row = r + 8*(lane/16), col = lane%16
  const int cb = N0 + wn * 64 + l15;
  #pragma unroll
  for (int r = 0; r < 8; ++r) {
    const int row0 = M0 + wm * 32 + r + hl * 8;
    float* c0 = C + (size_t)row0 * ldc + cb;
    float* c1 = C + (size_t)(row0 + 16) * ldc + cb;
    if (accum) {
      c0[0] += acc00[r]; c0[16] += acc01[r]; c0[32] += acc02[r]; c0[48] += acc03[r];
      c1[0] += acc10[r]; c1[16] += acc11[r]; c1[32] += acc12[r]; c1[48] += acc13[r];
    } else {
      c0[0]  = acc00[r]; c0[16]  = acc01[r]; c0[32]  = acc02[r]; c0[48]  = acc03[r];
      c1[0]  = acc10[r]; c1[16]  = acc11[r]; c1[32]  = acc12[r]; c1[48]  = acc13[r];
    }
  }
}

// ---------------- small elementwise / bias kernels ----------------
__global__ void prep_kernel(const float* __restrict__ act, const float* __restrict__ lns,
                            const float* __restrict__ pruned,
                            float* __restrict__ x_ln, float* __restrict__ up_ln,
                            float* __restrict__ inv_ls)
{
  const int bs = blockIdx.x;                 // 0..BS_-1
  const float inv = 1.0f / lns[bs];
  if (threadIdx.x == 0) inv_ls[bs] = inv;
  for (int d = threadIdx.x; d < D_; d += blockDim.x)
    x_ln[(size_t)bs * D_ + d] = act[(size_t)bs * D_ + d] * inv;
  for (int f = threadIdx.x; f < F_; f += blockDim.x)
    up_ln[(size_t)bs * F_ + f] = pruned[(size_t)bs * F_ + f] * inv;
}

// wov_ub[d] = sum_{h,m} W_OV[h,m,d] * up_b_dec[m]
__global__ void wov_ub_kernel(const float* __restrict__ W_OV,
                              const float* __restrict__ up_b_dec,
                              float* __restrict__ wovub)
{
  const int d = blockIdx.x * blockDim.x + threadIdx.x;
  if (d >= D_) return;
  float s = 0.0f;
  for (int h = 0; h < H_; ++h)
    for (int m = 0; m < D_; ++m)
      s += W_OV[((size_t)h * D_ + m) * D_ + d] * up_b_dec[m];
  wovub[d] = s;
}

// bias0[f] = b_enc[f] - W_enc[f,:]·b_dec ;  cf[f] = W_enc[f,:]·wov_ub
__global__ void biasf_kernel(const float* __restrict__ W_enc, const float* __restrict__ b_enc,
                             const float* __restrict__ b_dec, const float* __restrict__ wovub,
                             float* __restrict__ bias0, float* __restrict__ cf)
{
  const int f = blockIdx.x * blockDim.x + threadIdx.x;
  if (f >= F_) return;
  float s0 = 0.0f, s1 = 0.0f;
  for (int d = 0; d < D_; ++d) {
    const float w = W_enc[(size_t)f * D_ + d];
    s0 += w * b_dec[d];
    s1 += w * wovub[d];
  }
  bias0[f] = b_enc[f] - s0;
  cf[f]    = s1;
}

// out[bs,f] = bias0[f] + inv_ls[bs]*cf[f]  (GEMMs then accumulate on top)
__global__ void init_out_kernel(const float* __restrict__ inv_ls,
                                const float* __restrict__ bias0,
                                const float* __restrict__ cf,
                                float* __restrict__ out)
{
  const int bs = blockIdx.x;
  const float inv = inv_ls[bs];
  for (int f = threadIdx.x; f < F_; f += blockDim.x)
    out[(size_t)bs * F_ + f] = bias0[f] + inv * cf[f];
}

__global__ void zero_kernel(float* __restrict__ p, int n)
{
  const int i = blockIdx.x * blockDim.x + threadIdx.x;
  if (i < n) p[i] = 0.0f;
}

// vw *= connection_mask (jax bool -> 1 byte each)
__global__ void mask_kernel(float* __restrict__ vw, const unsigned char* __restrict__ m, int n)
{
  const int i = blockIdx.x * blockDim.x + threadIdx.x;
  if (i < n) vw[i] = m[i] ? vw[i] : 0.0f;
}

// ---------------- host orchestration ----------------
extern "C" void kernel_launch(void* const* d_in, const int* in_sizes, int n_in,
                              void* d_out, int out_size, void* d_ws, size_t ws_size,
                              hipStream_t stream)
{
  const float* initial_act = (const float*)d_in[0];
  const float* ln_scale    = (const float*)d_in[1];
  const float* probs       = (const float*)d_in[2];
  const float* pruned      = (const float*)d_in[3];
  const float* W_O         = (const float*)d_in[4];
  const float* W_V         = (const float*)d_in[5];
  const float* W_enc       = (const float*)d_in[6];
  const float* b_enc       = (const float*)d_in[7];
  const float* b_dec       = (const float*)d_in[8];
  const float* W_dec       = (const float*)d_in[9];
  const float* up_b_dec    = (const float*)d_in[10];
  const unsigned char* cmask = (const unsigned char*)d_in[11];
  float* out = (float*)d_out;

  // workspace partition (~104 MB of f32)
  float* p = (float*)d_ws;
  float* W_OV  = p; p += (size_t)H_ * D_ * D_;   // 7,077,888
  float* x_ln  = p; p += (size_t)BS_ * D_;       // 1,572,864
  float* up_ln = p; p += (size_t)BS_ * F_;       // 4,194,304
  float* inv_ls= p; p += BS_;                    // 2,048
  float* u     = p; p += (size_t)BS_ * D_;       // 1,572,864
  float* z     = p; p += (size_t)BS_ * D_;       // 1,572,864
  float* Ah    = p; p += (size_t)F_ * D_;        // 1,572,864
  float* vw    = p; p += (size_t)F_ * F_;        // 4,194,304
  float* rh    = p; p += (size_t)BS_ * F_;       // 4,194,304
  float* wovub = p; p += D_;
  float* bias0 = p; p += F_;
  float* cf    = p; p += F_;

  auto GEMM = [&](const float* A, const float* Bm, float* C,
                  int M, int N, int K, int lda, int ldb, int ldc,
                  int transB, int accum) {
    dim3 g((unsigned)(N / BN), (unsigned)(M / BM));
    gemm_f32_wmma<<<g, 256, 0, stream>>>(A, Bm, C, M, N, K, lda, ldb, ldc, transB, accum);
  };

  // 0) x_ln, up_ln, inv_ls
  prep_kernel<<<BS_, 256, 0, stream>>>(initial_act, ln_scale, pruned, x_ln, up_ln, inv_ls);

  // 1) W_OV[h] = W_V[h][D,DH] @ W_O[h][DH,D]
  for (int h = 0; h < H_; ++h)
    GEMM(W_V + (size_t)h * D_ * DH_, W_O + (size_t)h * DH_ * D_,
         W_OV + (size_t)h * D_ * D_, D_, D_, DH_, DH_, D_, D_, 0, 0);

  // 2) bias block collapsed to rank-1 update, written first so GEMMs accumulate on top
  wov_ub_kernel<<<(D_ + 255) / 256, 256, 0, stream>>>(W_OV, up_b_dec, wovub);
  biasf_kernel<<<(F_ + 255) / 256, 256, 0, stream>>>(W_enc, b_enc, b_dec, wovub, bias0, cf);
  init_out_kernel<<<BS_, 256, 0, stream>>>(inv_ls, bias0, cf, out);
  zero_kernel<<<(BS_ * D_ + 255) / 256, 256, 0, stream>>>(u, BS_ * D_);

  // 3) per-head pipeline (keeps vw/r at [F,F]/[B,S,F] instead of xH blowup)
  for (int h = 0; h < H_; ++h) {
    // z[b] = probs[b,h][S,S] @ x_ln[b][S,D]
    for (int b = 0; b < B_; ++b)
      GEMM(probs + ((size_t)b * H_ + h) * S_ * S_, x_ln + (size_t)b * S_ * D_,
           z + (size_t)b * S_ * D_, S_, D_, S_, S_, D_, D_, 0, 0);
    // u[b] += z[b] @ W_OV[h]
    for (int b = 0; b < B_; ++b)
      GEMM(z + (size_t)b * S_ * D_, W_OV + (size_t)h * D_ * D_,
           u + (size_t)b * S_ * D_, S_, D_, D_, D_, D_, D_, 0, 1);
    // Ah[f,m] = W_enc[F,D] @ W_OV[h]^T
    GEMM(W_enc, W_OV + (size_t)h * D_ * D_, Ah, F_, D_, D_, D_, D_, D_, 1, 0);
    // vw[f,g] = Ah @ W_dec[D,F], then mask
    GEMM(Ah, W_dec, vw, F_, F_, D_, D_, F_, F_, 0, 0);
    mask_kernel<<<(F_ * F_ + 255) / 256, 256, 0, stream>>>(vw, cmask, F_ * F_);
    // rh[b] = probs[b,h] @ up_ln[b][S,F]
    for (int b = 0; b < B_; ++b)
      GEMM(probs + ((size_t)b * H_ + h) * S_ * S_, up_ln + (size_t)b * S_ * F_,
           rh + (size_t)b * S_ * F_, S_, F_, S_, S_, F_, F_, 0, 0);
    // out[b] += rh[b] @ vw^T  (contraction over g)
    for (int b = 0; b < B_; ++b)
      GEMM(rh + (size_t)b * S_ * F_, vw, out + (size_t)b * S_ * F_,
           S_, F_, F_, F_, F_, F_, 1, 1);
  }

  // 4) initial contribs: out[b] += u[b][S,D] @ W_enc^T
  for (int b = 0; b < B_; ++b)
    GEMM(u + (size_t)b * S_ * D_, W_enc, out + (size_t)b * S_ * F_,
         S_, F_, D_, D_, D_, F_, 1, 1);
}